// yolov8_target_nose_mouth_83133386982264
// MI455X (gfx1250) — compile-verified
//
#include <hip/hip_runtime.h>
#include <hip/hip_bf16.h>

// yolov8 target-class masked score sum:
//   sum over rows of ( where(post[:,3] >= 0.2, post[:,3], 0) + where(post[:,4] >= 0.2, post[:,4], 0) )
// Memory-bound: ~1 cacheline fetched per 320B row -> ~128-256MB @ 23.3TB/s ~ 6-11us floor.

#define NUM_CLASSES 80
#define CONF_THRESH 0.2f
#define PART_BLOCKS 2048
#define PART_THREADS 256

typedef __attribute__((ext_vector_type(2))) float v2f;
typedef __attribute__((ext_vector_type(8))) float v8f;

__global__ void yolo_partial_sum_kernel(const float* __restrict__ post,
                                        float* __restrict__ partials,
                                        int n_rows) {
    __shared__ float red[PART_THREADS];
    const int tid    = blockIdx.x * blockDim.x + threadIdx.x;
    const int stride = gridDim.x * blockDim.x;

    float acc = 0.0f;
    for (int i = tid; i < n_rows; i += stride) {
        const float* row = post + (long long)i * NUM_CLASSES;
        float a = row[3];
        float b = row[4];
        acc += (a >= CONF_THRESH ? a : 0.0f);
        acc += (b >= CONF_THRESH ? b : 0.0f);
    }

    red[threadIdx.x] = acc;
    __syncthreads();
    for (int s = PART_THREADS / 2; s > 0; s >>= 1) {
        if (threadIdx.x < s) red[threadIdx.x] += red[threadIdx.x + s];
        __syncthreads();
    }
    if (threadIdx.x == 0) partials[blockIdx.x] = red[0];
}

// Single-wave (32 threads) final reduction. Cross-lane collapse is done with
// V_WMMA_F32_16X16X4_F32: A[16x4] carries the 32 lane accumulators
// (A[m,0] = acc(lane m), A[m,2] = acc(lane m+16), others 0), B = all ones,
// so D[m,n] = acc(m) + acc(m+16) for every column n. Each lane then sums its
// 8 C/D VGPRs (M=0..7 for lanes 0-15, M=8..15 for lanes 16-31) and one
// shfl_xor(16) produces the full 32-lane sum. EXEC is all 1s (full wave,
// no divergence at the WMMA), as the ISA requires.
__global__ void yolo_final_reduce_kernel(const float* __restrict__ partials,
                                         float* __restrict__ out,
                                         int n) {
    float acc = 0.0f;
    for (int i = threadIdx.x; i < n; i += 32) {
        acc += partials[i];
    }

    v2f a;
    a.x = acc;
    a.y = 0.0f;
    v2f b;
    b.x = 1.0f;
    b.y = 1.0f;
    v8f c = {};
    // (neg_a, A, neg_b, B, c_mod, C, reuse_a, reuse_b)
    c = __builtin_amdgcn_wmma_f32_16x16x4_f32(
        /*neg_a=*/false, a, /*neg_b=*/false, b,
        /*c_mod=*/(short)0, c, /*reuse_a=*/false, /*reuse_b=*/false);

    float s = c[0] + c[1] + c[2] + c[3] + c[4] + c[5] + c[6] + c[7];
    s += __shfl_xor(s, 16, 32);

    if (threadIdx.x == 0) {
        out[0] = s;
    }
}

extern "C" void kernel_launch(void* const* d_in, const int* in_sizes, int n_in,
                              void* d_out, int out_size, void* d_ws, size_t ws_size,
                              hipStream_t stream) {
    const float* post = (const float*)d_in[0];  // (n_rows, 80) f32
    // d_in[1] (pre_post_boxes) is unused by the reference output.
    float* out      = (float*)d_out;            // 1 element f32
    float* partials = (float*)d_ws;             // PART_BLOCKS floats of scratch

    const int n_rows = in_sizes[0] / NUM_CLASSES;

    yolo_partial_sum_kernel<<<PART_BLOCKS, PART_THREADS, 0, stream>>>(post, partials, n_rows);
    yolo_final_reduce_kernel<<<1, 32, 0, stream>>>(partials, out, PART_BLOCKS);
}